// HGTLayer_70849780514976
// MI455X (gfx1250) — compile-verified
//
#include <hip/hip_runtime.h>
#include <hip/hip_bf16.h>
#include <stdint.h>

// ---------------------------------------------------------------------------
// HGT layer forward for MI455X (gfx1250, wave32).
// - All GEMM-like stages: V_WMMA_F32_16X16X4_F32 (full f32; layer is
//   L2/gather bound, matrix units have huge headroom).
// - All LDS tile staging: GLOBAL_LOAD_ASYNC_TO_LDS_B128 (CDNA5 async DMA
//   path, ASYNCcnt), no VGPR round-trip.
// Workspace budget: ~246 MB.
// ---------------------------------------------------------------------------

constexpr int N_  = 50000;
constexpr int E_  = 400000;
constexpr int T_  = 4;
constexpr int R_  = 8;
constexpr int H_  = 8;
constexpr int DK_ = 32;
constexpr int DD  = 256;                      // DIN == DOUT == H*DK
constexpr float INV_SQRT_DK = 0.1767766953f;  // 1/sqrt(32)

typedef float v2f __attribute__((ext_vector_type(2)));
typedef float v8f __attribute__((ext_vector_type(8)));

// D = A(16x4) * B(4x16) + C(16x16), all f32.  (codegen-verified round 1)
//   A: lane l -> row M = l&15, VGPR0/1 = K = (l>>4)*2 + {0,1}
//   B: lane l -> col N = l&15, VGPR0/1 = K = (l>>4)*2 + {0,1}
//   C/D: lane l -> col N = l&15, VGPR i -> row M = i + 8*(l>>4)
static __device__ __forceinline__ v8f wmma_f32_4(v2f a, v2f b, v8f c) {
  return __builtin_amdgcn_wmma_f32_16x16x4_f32(
      /*neg_a=*/false, a, /*neg_b=*/false, b,
      /*c_mod=*/(short)0, c, /*reuse_a=*/false, /*reuse_b=*/false);
}

// --- CDNA5 async global->LDS DMA (ISA §15.18.3 op 98, tracked by ASYNCcnt) --
// GV mode: each lane copies 16 bytes from the global address in vaddr to the LDS address in vdst.
// LDS byte address = low 32 bits of the generic (flat) shared pointer: the
// LDS aperture keeps the workgroup-relative offset in addr[31:0].
static __device__ __forceinline__ void async_copy_b128(void* lds_dst,
                                                       const void* gsrc) {
  unsigned lds = (unsigned)(uintptr_t)lds_dst;
  asm volatile("global_load_async_to_lds_b128 %0, %1, off"
               :: "v"(lds), "v"(gsrc)
               : "memory");
}
static __device__ __forceinline__ void wait_async_all() {
  asm volatile("s_wait_asynccnt 0x0" ::: "memory");
}

// ---------------------------------------------------------------------------
// init: zero accumulators, reset counters, amax sentinel (monotone-uint 0).
// ---------------------------------------------------------------------------
__global__ void k_init(float* __restrict__ t_acc, float* __restrict__ denom,
                       unsigned* __restrict__ amax, int* __restrict__ ecnt,
                       int* __restrict__ ecur) {
  size_t i = (size_t)blockIdx.x * blockDim.x + threadIdx.x;
  size_t stride = (size_t)gridDim.x * blockDim.x;
  for (size_t j = i; j < (size_t)N_ * DD; j += stride) t_acc[j] = 0.f;
  for (size_t j = i; j < (size_t)N_ * R_ * H_; j += stride) {
    denom[j] = 0.f;
    amax[j] = 0u;
  }
  if (i < (size_t)R_) { ecnt[i] = 0; ecur[i] = 0; }
}

// ---------------------------------------------------------------------------
// Bucket edges by relation so WMMA tiles share one etype.
// ---------------------------------------------------------------------------
__global__ void k_hist(const int* __restrict__ etype, int* __restrict__ ecnt) {
  int i = blockIdx.x * blockDim.x + threadIdx.x;
  if (i < E_) atomicAdd(&ecnt[etype[i]], 1);
}

__global__ void k_scan(const int* __restrict__ ecnt, int* __restrict__ eoff,
                       int* __restrict__ ecur) {
  if (blockIdx.x == 0 && threadIdx.x == 0) {
    int acc = 0;
    for (int r = 0; r < R_; ++r) { eoff[r] = acc; ecur[r] = acc; acc += ecnt[r]; }
    eoff[R_] = acc;
  }
}

__global__ void k_scatter(const int* __restrict__ etype, int* __restrict__ ecur,
                          int* __restrict__ eperm) {
  int i = blockIdx.x * blockDim.x + threadIdx.x;
  if (i < E_) {
    int pos = atomicAdd(&ecur[etype[i]], 1);
    eperm[pos] = i;
  }
}

// ---------------------------------------------------------------------------
// Transpose rel_att / rel_msg (per r,h 32x32): raT[r,h,f,d] = ra[r,h,d,f],
// so B-operand loads (consecutive K at fixed output column) are float2s.
// ---------------------------------------------------------------------------
__global__ void k_transpose_rel(const float* __restrict__ ra,
                                const float* __restrict__ rm,
                                float* __restrict__ raT,
                                float* __restrict__ rmT) {
  int idx = blockIdx.x * blockDim.x + threadIdx.x;
  int total = R_ * H_ * DK_ * DK_;
  if (idx >= total) return;
  int f = idx & (DK_ - 1);
  int d = (idx >> 5) & (DK_ - 1);
  int rh = idx >> 10;
  int o = (rh * DK_ + f) * DK_ + d;
  raT[o] = ra[idx];
  rmT[o] = rm[idx];
}

// ---------------------------------------------------------------------------
// Typed K/Q/V projections: y_n = W[type_n] @ x_n + b[type_n].
// One block per 16-node tile, async-DMA x tile to LDS, compute for ALL 4
// types with predicated stores. 192 wave-jobs over 8 waves.
// ---------------------------------------------------------------------------
__global__ __launch_bounds__(256) void k_qkv(
    const float* __restrict__ x, const int* __restrict__ ntype,
    const float* __restrict__ Wk, const float* __restrict__ bk,
    const float* __restrict__ Wq, const float* __restrict__ bq,
    const float* __restrict__ Wv, const float* __restrict__ bv,
    float* __restrict__ Kb, float* __restrict__ Qb, float* __restrict__ Vb) {
  __shared__ float sx[16 * DD];
  __shared__ int stype[16];
  int nBase = blockIdx.x * 16;
  int tid = threadIdx.x;
  if (tid < 16) {
    int n = nBase + tid;
    stype[tid] = (n < N_) ? ntype[n] : -1;
  }
  // async DMA: 16 rows x 64 16B-chunks. Out-of-range rows left unissued:
  // their C rows are never stored (predicated below).
  for (int idx = tid; idx < 16 * 64; idx += 256) {
    int m = idx >> 6, cg = idx & 63;
    int n = nBase + m;
    if (n < N_)
      async_copy_b128(&sx[m * DD + cg * 4], x + (size_t)n * DD + cg * 4);
  }
  wait_async_all();
  __syncthreads();

  int w = tid >> 5, lane = tid & 31;
  int mr = lane & 15;              // A row == B col handled by this lane
  int koff = (lane >> 4) << 1;     // 0 or 2
  int rhalf = (lane >> 4) << 3;    // C row offset: 0 or 8

  for (int job = w; job < 3 * 4 * 16; job += 8) {
    int proj = job >> 6;
    int rem = job & 63;
    int t = rem >> 4;
    int dBase = (rem & 15) << 4;
    const float* Wsel = (proj == 0) ? Wk : ((proj == 1) ? Wq : Wv);
    const float* bsel = (proj == 0) ? bk : ((proj == 1) ? bq : bv);
    float* Obuf = (proj == 0) ? Kb : ((proj == 1) ? Qb : Vb);
    // B[k][n] = W[t][dBase+n][k]  -> contiguous in k
    const float* Wrow = Wsel + ((size_t)t * DD + (dBase + mr)) * DD;
    v8f acc = {};
    for (int k0 = 0; k0 < DD; k0 += 4) {
      v2f a = *(const v2f*)(sx + mr * DD + k0 + koff);
      v2f b = *(const v2f*)(Wrow + k0 + koff);
      acc = wmma_f32_4(a, b, acc);
    }
    float bias = bsel[t * DD + dBase + mr];
#pragma unroll
    for (int i = 0; i < 8; ++i) {
      int m = i + rhalf;
      int node = nBase + m;
      if (node < N_ && stype[m] == t)
        Obuf[(size_t)node * DD + dBase + mr] = acc[i] + bias;
    }
  }
}

// ---------------------------------------------------------------------------
// Attention scores: per 16-edge same-relation tile, per head h:
//   key = k_e(16x32) @ rel_att[r,h](32x32)   (WMMA, 2 col-tiles x 8 K-steps)
//   att[e,h] = dot(q_e[h], key[h]) * rel_pri[r,h] / sqrt(DK)
// then monotone-uint atomicMax into amax[(dst*R+r)*H + h].
// K/Q rows (L2-resident) gathered into LDS via async DMA.
// ---------------------------------------------------------------------------
__global__ __launch_bounds__(256) void k_att(
    const float* __restrict__ Kb, const float* __restrict__ Qb,
    const int* __restrict__ src, const int* __restrict__ dst,
    const int* __restrict__ eoff, const int* __restrict__ eperm,
    const float* __restrict__ RattT, const float* __restrict__ rel_pri,
    float* __restrict__ attbuf, unsigned* __restrict__ amax) {
  __shared__ float sk[16 * DD];
  __shared__ float sq[16 * DD];
  __shared__ float skey[H_][16][DK_];
  __shared__ int se[16];
  __shared__ int ssrc[16];
  __shared__ int sdst[16];

  int r = blockIdx.y;
  int start = eoff[r] + blockIdx.x * 16;
  int end = eoff[r + 1];
  if (start >= end) return;
  int count = min(16, end - start);
  int tid = threadIdx.x;

  if (tid < 16) {
    int e = (tid < count) ? eperm[start + tid] : -1;
    se[tid] = e;
    ssrc[tid] = (e >= 0) ? src[e] : 0;
    sdst[tid] = (e >= 0) ? dst[e] : 0;
  }
  __syncthreads();
  // gather k/q rows: async DMA, 2 x (16 rows x 64 chunks)
  for (int idx = tid; idx < 16 * 64; idx += 256) {
    int m = idx >> 6, cg = idx & 63;
    if (se[m] >= 0) {
      async_copy_b128(&sk[m * DD + cg * 4],
                      Kb + (size_t)ssrc[m] * DD + cg * 4);
      async_copy_b128(&sq[m * DD + cg * 4],
                      Qb + (size_t)sdst[m] * DD + cg * 4);
    }
  }
  wait_async_all();
  __syncthreads();

  int w = tid >> 5, lane = tid & 31;
  int mr = lane & 15;
  int koff = (lane >> 4) << 1;
  int rhalf = (lane >> 4) << 3;
  int h = w;  // 8 waves -> 8 heads
  const float* Bt = RattT + ((size_t)r * H_ + h) * DK_ * DK_;  // [f][d]
  for (int ct = 0; ct < 2; ++ct) {
    int cb = ct << 4;
    const float* Brow = Bt + (cb + mr) * DK_;
    v8f acc = {};
    for (int k0 = 0; k0 < DK_; k0 += 4) {
      v2f a = *(const v2f*)(sk + mr * DD + h * DK_ + k0 + koff);
      v2f b = *(const v2f*)(Brow + k0 + koff);
      acc = wmma_f32_4(a, b, acc);
    }
#pragma unroll
    for (int i = 0; i < 8; ++i) skey[h][i + rhalf][cb + mr] = acc[i];
  }
  __syncthreads();

  if (tid < 16 * H_) {
    int m = tid >> 3;
    int hh = tid & 7;
    if (m < count) {
      float s = 0.f;
#pragma unroll
      for (int f = 0; f < DK_; ++f)
        s += sq[m * DD + hh * DK_ + f] * skey[hh][m][f];
      s = s * rel_pri[r * H_ + hh] * INV_SQRT_DK;
      attbuf[(size_t)(start + m) * H_ + hh] = s;
      unsigned ub = __float_as_uint(s);
      unsigned u = (s >= 0.f) ? (ub | 0x80000000u) : ~ub;
      atomicMax(&amax[(size_t)(sdst[m] * R_ + r) * H_ + hh], u);
    }
  }
}

// ---------------------------------------------------------------------------
// exp pass: ex = exp(att - amax[seg]); attbuf <- ex; denom[seg] += ex.
// ---------------------------------------------------------------------------
__global__ void k_ex(const int* __restrict__ dst, const int* __restrict__ etype,
                     const int* __restrict__ eperm, float* __restrict__ attbuf,
                     const unsigned* __restrict__ amax,
                     float* __restrict__ denom) {
  int idx = blockIdx.x * blockDim.x + threadIdx.x;
  if (idx >= E_ * H_) return;
  int p = idx >> 3, h = idx & 7;
  int e = eperm[p];
  int seg = dst[e] * R_ + etype[e];
  unsigned u = amax[(size_t)seg * H_ + h];
  float mx = (u & 0x80000000u) ? __uint_as_float(u & 0x7FFFFFFFu)
                               : __uint_as_float(~u);
  float ex = __expf(attbuf[idx] - mx);
  attbuf[idx] = ex;
  atomicAdd(&denom[(size_t)seg * H_ + h], ex);
}

// ---------------------------------------------------------------------------
// Message pass: per 16-edge same-relation tile, per head:
//   msg = v_e(16x32) @ rel_msg[r,h](32x32)   (WMMA)
//   t_acc[dst] += (ex/denom) * msg           (atomic scatter into 51MB
//                                             L2-resident accumulator)
// Summing per-segment-normalized contributions directly into the node
// accumulator equals the reference's [N,R,D] segment buffer summed over R.
// ---------------------------------------------------------------------------
__global__ __launch_bounds__(256) void k_msg(
    const float* __restrict__ Vb, const int* __restrict__ src,
    const int* __restrict__ dst, const int* __restrict__ eoff,
    const int* __restrict__ eperm, const float* __restrict__ RmsgT,
    const float* __restrict__ attbuf, const float* __restrict__ denom,
    float* __restrict__ t_acc) {
  __shared__ float sv[16 * DD];
  __shared__ float sw[16][H_];
  __shared__ int se[16];
  __shared__ int ssrc[16];
  __shared__ int sdst[16];

  int r = blockIdx.y;
  int start = eoff[r] + blockIdx.x * 16;
  int end = eoff[r + 1];
  if (start >= end) return;
  int count = min(16, end - start);
  int tid = threadIdx.x;

  if (tid < 16) {
    int e = (tid < count) ? eperm[start + tid] : -1;
    se[tid] = e;
    ssrc[tid] = (e >= 0) ? src[e] : 0;
    sdst[tid] = (e >= 0) ? dst[e] : 0;
  }
  __syncthreads();
  for (int idx = tid; idx < 16 * 64; idx += 256) {
    int m = idx >> 6, cg = idx & 63;
    if (se[m] >= 0)
      async_copy_b128(&sv[m * DD + cg * 4],
                      Vb + (size_t)ssrc[m] * DD + cg * 4);
  }
  if (tid < 16 * H_) {
    int m = tid >> 3, h = tid & 7;
    float wgt = 0.f;
    if (se[m] >= 0) {
      int seg = sdst[m] * R_ + r;
      float d = denom[(size_t)seg * H_ + h];
      wgt = (d > 0.f) ? attbuf[(size_t)(start + m) * H_ + h] / d : 0.f;
    }
    sw[m][h] = wgt;
  }
  wait_async_all();
  __syncthreads();

  int w = tid >> 5, lane = tid & 31;
  int mr = lane & 15;
  int koff = (lane >> 4) << 1;
  int rhalf = (lane >> 4) << 3;
  int h = w;
  const float* Bt = RmsgT + ((size_t)r * H_ + h) * DK_ * DK_;
  for (int ct = 0; ct < 2; ++ct) {
    int cb = ct << 4;
    const float* Brow = Bt + (cb + mr) * DK_;
    v8f acc = {};
    for (int k0 = 0; k0 < DK_; k0 += 4) {
      v2f a = *(const v2f*)(sv + mr * DD + h * DK_ + k0 + koff);
      v2f b = *(const v2f*)(Brow + k0 + koff);
      acc = wmma_f32_4(a, b, acc);
    }
#pragma unroll
    for (int i = 0; i < 8; ++i) {
      int m = i + rhalf;
      if (m < count) {
        float val = acc[i] * sw[m][h];
        atomicAdd(&t_acc[(size_t)sdst[m] * DD + h * DK_ + cb + mr], val);
      }
    }
  }
}

// ---------------------------------------------------------------------------
// Output: t = t_acc / max(#present etypes,1); trans = Wa[type] @ t + ba;
// out = trans*sigmoid(skip[type]) + x*(1-sigmoid).
// t rows async-DMA'd to LDS then scaled in place.
// ---------------------------------------------------------------------------
__global__ __launch_bounds__(256) void k_out(
    const float* __restrict__ x, const int* __restrict__ ntype,
    const float* __restrict__ t_acc, const float* __restrict__ denom,
    const float* __restrict__ Wa, const float* __restrict__ ba,
    const float* __restrict__ skip, float* __restrict__ out) {
  __shared__ float st[16 * DD];
  __shared__ float salpha[16];
  __shared__ float sinv[16];
  __shared__ int stype[16];
  int nBase = blockIdx.x * 16;
  int tid = threadIdx.x;
  if (tid < 16) {
    int n = nBase + tid;
    int ty = (n < N_) ? ntype[n] : 0;
    stype[tid] = (n < N_) ? ty : -1;
    salpha[tid] = (n < N_) ? 1.f / (1.f + __expf(-skip[ty])) : 0.f;
  } else if (tid < 32) {
    int m = tid - 16;
    int n = nBase + m;
    int cnt = 0;
    if (n < N_)
      for (int r = 0; r < R_; ++r)
        cnt += (denom[((size_t)n * R_ + r) * H_] > 0.f) ? 1 : 0;
    sinv[m] = 1.f / fmaxf((float)cnt, 1.f);
  }
  for (int idx = tid; idx < 16 * 64; idx += 256) {
    int m = idx >> 6, cg = idx & 63;
    int n = nBase + m;
    if (n < N_)
      async_copy_b128(&st[m * DD + cg * 4],
                      t_acc + (size_t)n * DD + cg * 4);
  }
  wait_async_all();
  __syncthreads();
  // scale by 1/presence count (in place; invalid rows unused downstream)
  for (int idx = tid; idx < 16 * DD; idx += 256) {
    int m = idx >> 8;
    int n = nBase + m;
    if (n < N_) st[idx] *= sinv[m];
  }
  __syncthreads();

  int w = tid >> 5, lane = tid & 31;
  int mr = lane & 15;
  int koff = (lane >> 4) << 1;
  int rhalf = (lane >> 4) << 3;
  for (int job = w; job < 4 * 16; job += 8) {
    int t = job >> 4;
    int dBase = (job & 15) << 4;
    const float* Wrow = Wa + ((size_t)t * DD + (dBase + mr)) * DD;
    v8f acc = {};
    for (int k0 = 0; k0 < DD; k0 += 4) {
      v2f a = *(const v2f*)(st + mr * DD + k0 + koff);
      v2f b = *(const v2f*)(Wrow + k0 + koff);
      acc = wmma_f32_4(a, b, acc);
    }
    float bias = ba[t * DD + dBase + mr];
#pragma unroll
    for (int i = 0; i < 8; ++i) {
      int m = i + rhalf;
      int node = nBase + m;
      if (node < N_ && stype[m] == t) {
        float trans = acc[i] + bias;
        float al = salpha[m];
        size_t o = (size_t)node * DD + dBase + mr;
        out[o] = trans * al + x[o] * (1.f - al);
      }
    }
  }
}

// ---------------------------------------------------------------------------
extern "C" void kernel_launch(void* const* d_in, const int* in_sizes, int n_in,
                              void* d_out, int out_size, void* d_ws,
                              size_t ws_size, hipStream_t stream) {
  (void)in_sizes; (void)n_in; (void)out_size; (void)ws_size;
  const float* x        = (const float*)d_in[0];
  const int*   node_type= (const int*)d_in[1];
  const int*   src      = (const int*)d_in[2];
  const int*   dst      = (const int*)d_in[3];
  const int*   etype    = (const int*)d_in[4];
  const float* Wk = (const float*)d_in[5];
  const float* bk = (const float*)d_in[6];
  const float* Wq = (const float*)d_in[7];
  const float* bq = (const float*)d_in[8];
  const float* Wv = (const float*)d_in[9];
  const float* bv = (const float*)d_in[10];
  const float* Wa = (const float*)d_in[11];
  const float* ba = (const float*)d_in[12];
  const float* rel_pri = (const float*)d_in[13];
  const float* rel_att = (const float*)d_in[14];
  const float* rel_msg = (const float*)d_in[15];
  const float* skip    = (const float*)d_in[16];
  float* out = (float*)d_out;

  char* p = (char*)d_ws;
  auto alloc = [&](size_t bytes) -> void* {
    void* q = (void*)p;
    p += (bytes + 255) & ~(size_t)255;
    return q;
  };
  float*    Kb     = (float*)alloc((size_t)N_ * DD * 4);
  float*    Qb     = (float*)alloc((size_t)N_ * DD * 4);
  float*    Vb     = (float*)alloc((size_t)N_ * DD * 4);
  float*    t_acc  = (float*)alloc((size_t)N_ * DD * 4);
  float*    attbuf = (float*)alloc((size_t)E_ * H_ * 4);
  unsigned* amax   = (unsigned*)alloc((size_t)N_ * R_ * H_ * 4);
  float*    denom  = (float*)alloc((size_t)N_ * R_ * H_ * 4);
  float*    RattT  = (float*)alloc((size_t)R_ * H_ * DK_ * DK_ * 4);
  float*    RmsgT  = (float*)alloc((size_t)R_ * H_ * DK_ * DK_ * 4);
  int*      ecnt   = (int*)alloc(R_ * 4);
  int*      eoff   = (int*)alloc((R_ + 1) * 4);
  int*      ecur   = (int*)alloc(R_ * 4);
  int*      eperm  = (int*)alloc((size_t)E_ * 4);

  k_init<<<4096, 256, 0, stream>>>(t_acc, denom, amax, ecnt, ecur);
  k_hist<<<(E_ + 255) / 256, 256, 0, stream>>>(etype, ecnt);
  k_scan<<<1, 1, 0, stream>>>(ecnt, eoff, ecur);
  k_scatter<<<(E_ + 255) / 256, 256, 0, stream>>>(etype, ecur, eperm);
  k_transpose_rel<<<(R_ * H_ * DK_ * DK_ + 255) / 256, 256, 0, stream>>>(
      rel_att, rel_msg, RattT, RmsgT);
  k_qkv<<<(N_ + 15) / 16, 256, 0, stream>>>(x, node_type, Wk, bk, Wq, bq, Wv,
                                            bv, Kb, Qb, Vb);
  dim3 gE((E_ + 15) / 16, R_);
  k_att<<<gE, 256, 0, stream>>>(Kb, Qb, src, dst, eoff, eperm, RattT, rel_pri,
                                attbuf, amax);
  k_ex<<<(E_ * H_ + 255) / 256, 256, 0, stream>>>(dst, etype, eperm, attbuf,
                                                  amax, denom);
  k_msg<<<gE, 256, 0, stream>>>(Vb, src, dst, eoff, eperm, RmsgT, attbuf,
                                denom, t_acc);
  k_out<<<(N_ + 15) / 16, 256, 0, stream>>>(x, node_type, t_acc, denom, Wa, ba,
                                            skip, out);
}